// InnerPatchSoftShiftTripleModule_50946902065197
// MI455X (gfx1250) — compile-verified
//
#include <hip/hip_runtime.h>
#include <math.h>

typedef __attribute__((ext_vector_type(2))) float v2f;
typedef __attribute__((ext_vector_type(8))) float v8f;

static constexpr int Bn   = 2;
static constexpr int Hh   = 48;
static constexpr int Ww   = 48;
static constexpr int Lm   = Hh * Ww;          // 2304
static constexpr long L2m = (long)Lm * Lm;    // 5,308,416
static constexpr int C2   = 256;              // half channels
static constexpr int CIN  = 512;
static constexpr int COUT = 768;

// ---------------------------------------------------------------------------
// copy input (former+latter) into first 512 output channels per batch
// ---------------------------------------------------------------------------
__global__ void copyin_kernel(const float* __restrict__ in, float* __restrict__ out) {
    long idx = (long)blockIdx.x * blockDim.x + threadIdx.x;
    long n = (long)Bn * CIN * Lm;
    if (idx >= n) return;
    int b = (int)(idx / ((long)CIN * Lm));
    long r = idx - (long)b * CIN * Lm;
    out[(long)b * COUT * Lm + r] = in[idx];
}

// ---------------------------------------------------------------------------
// sq[b,l] = sum_c latter[b,c,l]^2
// ---------------------------------------------------------------------------
__global__ void sq_kernel(const float* __restrict__ in, float* __restrict__ sq) {
    int idx = blockIdx.x * blockDim.x + threadIdx.x;
    if (idx >= Bn * Lm) return;
    int b = idx / Lm, l = idx - b * Lm;
    const float* lat = in + (long)b * CIN * Lm + (long)C2 * Lm;
    float s = 0.f;
    for (int c = 0; c < C2; ++c) {
        float v = lat[(long)c * Lm + l];
        s += v * v;
    }
    sq[idx] = s;
}

// ---------------------------------------------------------------------------
// den[b,l] = sqrt( sum_{o in 3x3, 2D-bounds} sq[b, l+o] )
// mm[l]    = (mean of 3x3 mask patch <= thred/9) ? 1 : 0     (b==0 only)
// ---------------------------------------------------------------------------
__global__ void prep_kernel(const float* __restrict__ sq, const float* __restrict__ mask,
                            const int* __restrict__ mask_thred,
                            float* __restrict__ den, float* __restrict__ mmv) {
    int idx = blockIdx.x * blockDim.x + threadIdx.x;
    if (idx >= Bn * Lm) return;
    int b = idx / Lm, l = idx - b * Lm;
    int y = l / Ww, x = l - y * Ww;
    float s = 0.f, ms = 0.f;
    for (int dy = -1; dy <= 1; ++dy)
        for (int dx = -1; dx <= 1; ++dx) {
            int yy = y + dy, xx = x + dx;
            if (yy >= 0 && yy < Hh && xx >= 0 && xx < Ww) {
                int ll = yy * Ww + xx;
                s += sq[b * Lm + ll];
                if (b == 0) ms += mask[ll];
            }
        }
    den[idx] = sqrtf(s);
    if (b == 0) {
        float thr = (float)mask_thred[0] / 9.0f;
        mmv[l] = (ms / 9.0f <= thr) ? 1.0f : 0.0f;
    }
}

// ---------------------------------------------------------------------------
// GEMM1 (WMMA f32): G[b,p,l] = sum_c former[b,c,p] * latter[b,c,l]
//   A: M=p, K=c  (former, layout [c][p] -> K-major rows, M contiguous)
//   B: K=c, N=l  (latter, layout [c][l] -> N contiguous)
// wave: 32x32 tile (2x2 WMMA);  block: 8 waves -> 64(M) x 128(N)
// ---------------------------------------------------------------------------
__global__ void gemm_corr_kernel(const float* __restrict__ in, float* __restrict__ G) {
    const int b = blockIdx.z;
    const float* Ab = in + (long)b * CIN * Lm;          // former [c][p]
    const float* Bb = Ab + (long)C2 * Lm;               // latter [c][l]
    float* Gb = G + (long)b * L2m;

    const int wave = threadIdx.x >> 5;
    const int lane = threadIdx.x & 31;
    const int half = lane >> 4;
    const int lidx = lane & 15;
    const int Mbase = blockIdx.x * 64  + (wave >> 2) * 32;
    const int Nbase = blockIdx.y * 128 + (wave & 3) * 32;

    v8f acc[2][2] = {};
    for (int k = 0; k < C2; k += 4) {
        const int kk = k + 2 * half;
        v2f a[2], bv[2];
#pragma unroll
        for (int mi = 0; mi < 2; ++mi) {
            const int m = Mbase + mi * 16 + lidx;
            a[mi].x = Ab[(long)kk * Lm + m];
            a[mi].y = Ab[(long)(kk + 1) * Lm + m];
        }
#pragma unroll
        for (int ni = 0; ni < 2; ++ni) {
            const int n = Nbase + ni * 16 + lidx;
            bv[ni].x = Bb[(long)kk * Lm + n];
            bv[ni].y = Bb[(long)(kk + 1) * Lm + n];
        }
#pragma unroll
        for (int mi = 0; mi < 2; ++mi)
#pragma unroll
            for (int ni = 0; ni < 2; ++ni)
                acc[mi][ni] = __builtin_amdgcn_wmma_f32_16x16x4_f32(
                    false, a[mi], false, bv[ni], (short)0, acc[mi][ni], false, false);
    }
#pragma unroll
    for (int mi = 0; mi < 2; ++mi)
#pragma unroll
        for (int ni = 0; ni < 2; ++ni) {
            const int n = Nbase + ni * 16 + lidx;
#pragma unroll
            for (int r = 0; r < 8; ++r) {
                const int m = Mbase + mi * 16 + half * 8 + r;
                Gb[(long)m * Lm + n] = acc[mi][ni][r];
            }
        }
}

// ---------------------------------------------------------------------------
// scores: Y[b,l,p] = ( sum_{o, 2D-bounds both} G[b, p+o, l+o] ) / den[b,l]
// ---------------------------------------------------------------------------
__global__ void score_kernel(const float* __restrict__ G, const float* __restrict__ den,
                             float* __restrict__ Y) {
    long idx = (long)blockIdx.x * blockDim.x + threadIdx.x;
    if (idx >= (long)Bn * L2m) return;
    int b = (int)(idx / L2m);
    long rem = idx - (long)b * L2m;
    int l = (int)(rem / Lm), p = (int)(rem - (long)l * Lm);
    int ly = l / Ww, lx = l - ly * Ww;
    int py = p / Ww, px = p - py * Ww;
    const float* Gb = G + (long)b * L2m;
    float s = 0.f;
    for (int dy = -1; dy <= 1; ++dy)
        for (int dx = -1; dx <= 1; ++dx) {
            int ly2 = ly + dy, lx2 = lx + dx, py2 = py + dy, px2 = px + dx;
            if (ly2 >= 0 && ly2 < Hh && lx2 >= 0 && lx2 < Ww &&
                py2 >= 0 && py2 < Hh && px2 >= 0 && px2 < Ww)
                s += Gb[(long)(py2 * Ww + px2) * Lm + (ly2 * Ww + lx2)];
        }
    Y[idx] = s / den[b * Lm + l];
}

// ---------------------------------------------------------------------------
// fuse pass 1 (y-major flattened diagonal, flat-index bounds):
//   T[b,l,p] = sum_{d=-1..1, 0<=l+d<L, 0<=p+d<L} Y[b,l+d,p+d]
// ---------------------------------------------------------------------------
__global__ void fuse1_kernel(const float* __restrict__ Y, float* __restrict__ T) {
    long idx = (long)blockIdx.x * blockDim.x + threadIdx.x;
    if (idx >= (long)Bn * L2m) return;
    int b = (int)(idx / L2m);
    long rem = idx - (long)b * L2m;
    int l = (int)(rem / Lm), p = (int)(rem - (long)l * Lm);
    const float* Yb = Y + (long)b * L2m;
    float s = 0.f;
#pragma unroll
    for (int d = -1; d <= 1; ++d) {
        int ld = l + d, pd = p + d;
        if (ld >= 0 && ld < Lm && pd >= 0 && pd < Lm)
            s += Yb[(long)ld * Lm + pd];
    }
    T[idx] = s;
}

// ---------------------------------------------------------------------------
// fuse pass 2 (x-major flattened diagonal) + multiply by mm[l]:
//   m(l) = (l%W)*H + l/W ;  inverse: flat(m) = (m%H)*W + m/H
//   Y[b,l,p] = mm[l] * sum_{d, 0<=m(l)+d<L, 0<=m(p)+d<L} T[b, inv(m(l)+d), inv(m(p)+d)]
// ---------------------------------------------------------------------------
__global__ void fuse2_kernel(const float* __restrict__ T, const float* __restrict__ mmv,
                             float* __restrict__ Y) {
    long idx = (long)blockIdx.x * blockDim.x + threadIdx.x;
    if (idx >= (long)Bn * L2m) return;
    int b = (int)(idx / L2m);
    long rem = idx - (long)b * L2m;
    int l = (int)(rem / Lm), p = (int)(rem - (long)l * Lm);
    int ml = (l % Ww) * Hh + (l / Ww);
    int mp = (p % Ww) * Hh + (p / Ww);
    const float* Tb = T + (long)b * L2m;
    float s = 0.f;
#pragma unroll
    for (int d = -1; d <= 1; ++d) {
        int a = ml + d, c = mp + d;
        if (a >= 0 && a < Lm && c >= 0 && c < Lm) {
            int l2 = (a % Hh) * Ww + (a / Hh);
            int p2 = (c % Hh) * Ww + (c / Hh);
            s += Tb[(long)l2 * Lm + p2];
        }
    }
    Y[idx] = s * mmv[l];
}

// ---------------------------------------------------------------------------
// column-wise softmax over l (rows) for each p, then *mm[l].  In-place on Y.
// block = 256 threads = 8 row-groups x 32 columns
// ---------------------------------------------------------------------------
__global__ void softmax_kernel(float* __restrict__ Y, const float* __restrict__ mmv) {
    __shared__ float red[8 * 32];
    __shared__ float colv[32];
    const int pl = threadIdx.x & 31;
    const int g  = threadIdx.x >> 5;
    const int p  = blockIdx.x * 32 + pl;
    float* Yb = Y + (long)blockIdx.y * L2m;

    float m = -__builtin_huge_valf();
    for (int l = g; l < Lm; l += 8) m = fmaxf(m, Yb[(long)l * Lm + p]);
    red[g * 32 + pl] = m;
    __syncthreads();
    if (g == 0) {
        float mx = red[pl];
        for (int i = 1; i < 8; ++i) mx = fmaxf(mx, red[i * 32 + pl]);
        colv[pl] = mx;
    }
    __syncthreads();
    m = colv[pl];

    float s = 0.f;
    for (int l = g; l < Lm; l += 8) {
        long o = (long)l * Lm + p;
        float e = __expf(Yb[o] - m);
        Yb[o] = e;
        s += e;
    }
    red[g * 32 + pl] = s;
    __syncthreads();
    if (g == 0) {
        float ss = red[pl];
        for (int i = 1; i < 8; ++i) ss += red[i * 32 + pl];
        colv[pl] = ss;
    }
    __syncthreads();
    float inv = 1.0f / colv[pl];
    for (int l = g; l < Lm; l += 8) {
        long o = (long)l * Lm + p;
        Yb[o] = Yb[o] * inv * mmv[l];
    }
}

// ---------------------------------------------------------------------------
// Ecos[b,l,p] = sum_{o, 2D-bounds both} cosine[b, l+o, p+o]   (Y -> G)
// ---------------------------------------------------------------------------
__global__ void ecos_kernel(const float* __restrict__ Y, float* __restrict__ G) {
    long idx = (long)blockIdx.x * blockDim.x + threadIdx.x;
    if (idx >= (long)Bn * L2m) return;
    int b = (int)(idx / L2m);
    long rem = idx - (long)b * L2m;
    int l = (int)(rem / Lm), p = (int)(rem - (long)l * Lm);
    int ly = l / Ww, lx = l - ly * Ww;
    int py = p / Ww, px = p - py * Ww;
    const float* Yb = Y + (long)b * L2m;
    float s = 0.f;
    for (int dy = -1; dy <= 1; ++dy)
        for (int dx = -1; dx <= 1; ++dx) {
            int ly2 = ly + dy, lx2 = lx + dx, py2 = py + dy, px2 = px + dx;
            if (ly2 >= 0 && ly2 < Hh && lx2 >= 0 && lx2 < Ww &&
                py2 >= 0 && py2 < Hh && px2 >= 0 && px2 < Ww)
                s += Yb[(long)(ly2 * Ww + lx2) * Lm + (py2 * Ww + px2)];
        }
    G[idx] = s;
}

// ---------------------------------------------------------------------------
// GEMM2 (WMMA f32): shift[b,c,p] = (sum_l latter[b,c,l] * E[b,l,p]) / 9
//   A: M=c, K=l (latter [c][l], K contiguous)
//   B: K=l, N=p (E [l][p], N contiguous)
// writes to output channels 512..767
// ---------------------------------------------------------------------------
__global__ void gemm_agg_kernel(const float* __restrict__ in, const float* __restrict__ E,
                                float* __restrict__ out) {
    const int b = blockIdx.z;
    const float* Ab = in + (long)b * CIN * Lm + (long)C2 * Lm;   // latter [c][l]
    const float* Bb = E + (long)b * L2m;                         // E [l][p]
    float* Ob = out + (long)b * COUT * Lm + (long)CIN * Lm;

    const int wave = threadIdx.x >> 5;
    const int lane = threadIdx.x & 31;
    const int half = lane >> 4;
    const int lidx = lane & 15;
    const int Mbase = blockIdx.x * 64  + (wave >> 2) * 32;
    const int Nbase = blockIdx.y * 128 + (wave & 3) * 32;

    v8f acc[2][2] = {};
    for (int k = 0; k < Lm; k += 4) {
        const int kk = k + 2 * half;
        v2f a[2], bv[2];
#pragma unroll
        for (int mi = 0; mi < 2; ++mi) {
            const int m = Mbase + mi * 16 + lidx;
            a[mi].x = Ab[(long)m * Lm + kk];
            a[mi].y = Ab[(long)m * Lm + kk + 1];
        }
#pragma unroll
        for (int ni = 0; ni < 2; ++ni) {
            const int n = Nbase + ni * 16 + lidx;
            bv[ni].x = Bb[(long)kk * Lm + n];
            bv[ni].y = Bb[(long)(kk + 1) * Lm + n];
        }
#pragma unroll
        for (int mi = 0; mi < 2; ++mi)
#pragma unroll
            for (int ni = 0; ni < 2; ++ni)
                acc[mi][ni] = __builtin_amdgcn_wmma_f32_16x16x4_f32(
                    false, a[mi], false, bv[ni], (short)0, acc[mi][ni], false, false);
    }
    const float inv9 = 1.0f / 9.0f;
#pragma unroll
    for (int mi = 0; mi < 2; ++mi)
#pragma unroll
        for (int ni = 0; ni < 2; ++ni) {
            const int n = Nbase + ni * 16 + lidx;
#pragma unroll
            for (int r = 0; r < 8; ++r) {
                const int m = Mbase + mi * 16 + half * 8 + r;
                Ob[(long)m * Lm + n] = acc[mi][ni][r] * inv9;
            }
        }
}

// ---------------------------------------------------------------------------
extern "C" void kernel_launch(void* const* d_in, const int* in_sizes, int n_in,
                              void* d_out, int out_size, void* d_ws, size_t ws_size,
                              hipStream_t stream) {
    const float* input      = (const float*)d_in[0];
    const float* mask       = (const float*)d_in[1];
    const int*   mask_thred = (const int*)d_in[4];
    float* out = (float*)d_out;

    // workspace carving (floats): G | Y | sq | den | mm  (~85 MB total)
    float* G   = (float*)d_ws;
    float* Y   = G + (long)Bn * L2m;
    float* sq  = Y + (long)Bn * L2m;
    float* den = sq + Bn * Lm;
    float* mmv = den + Bn * Lm;

    const int TB = 256;
    long nCopy = (long)Bn * CIN * Lm;
    copyin_kernel<<<(int)((nCopy + TB - 1) / TB), TB, 0, stream>>>(input, out);

    int nBL = Bn * Lm;
    sq_kernel<<<(nBL + TB - 1) / TB, TB, 0, stream>>>(input, sq);
    prep_kernel<<<(nBL + TB - 1) / TB, TB, 0, stream>>>(sq, mask, mask_thred, den, mmv);

    // GEMM1: G[b,p,l]   (M=2304 -> 36 blocks of 64, N=2304 -> 18 blocks of 128)
    gemm_corr_kernel<<<dim3(36, 18, Bn), TB, 0, stream>>>(input, G);

    long nE = (long)Bn * L2m;
    int nEB = (int)((nE + TB - 1) / TB);
    score_kernel<<<nEB, TB, 0, stream>>>(G, den, Y);   // G -> Y (scores / den)
    fuse1_kernel<<<nEB, TB, 0, stream>>>(Y, G);        // Y -> G
    fuse2_kernel<<<nEB, TB, 0, stream>>>(G, mmv, Y);   // G -> Y (+ mask)

    softmax_kernel<<<dim3(Lm / 32, Bn), TB, 0, stream>>>(Y, mmv);  // in-place

    ecos_kernel<<<nEB, TB, 0, stream>>>(Y, G);         // Y -> G (Ecos)

    // GEMM2: shift (M=256 -> 4 blocks of 64, N=2304 -> 18 blocks of 128)
    gemm_agg_kernel<<<dim3(4, 18, Bn), TB, 0, stream>>>(input, G, out);
}